// DAODetector_41721312313533
// MI455X (gfx1250) — compile-verified
//
#include <hip/hip_runtime.h>
#include <hip/hip_bf16.h>
#include <math.h>

typedef __attribute__((ext_vector_type(2))) float v2f;
typedef __attribute__((ext_vector_type(8))) float v8f;

constexpr int kN    = 8192;   // points
constexpr int kD    = 256;    // feature dim
constexpr int kK    = 32;     // K in reference
constexpr int kROWS = 32;     // rows per workgroup
constexpr int kCT   = 64;     // columns per tile iteration
constexpr int kNSEG = 4;      // selection segments (== waves per WG)
constexpr int kLIST = 34;     // K+2 smallest (incl. self-zero)
#define kEPS 1e-4f

// ---------------- row squared norms ----------------
__global__ __launch_bounds__(256) void dao_norms(const float* __restrict__ X,
                                                 float* __restrict__ x2) {
  int lane = threadIdx.x & 31;
  int row  = blockIdx.x * 8 + (threadIdx.x >> 5);
  const float* p = X + (size_t)row * kD + lane * 8;
  float s = 0.f;
#pragma unroll
  for (int t = 0; t < 8; ++t) { float v = p[t]; s += v * v; }
#pragma unroll
  for (int off = 16; off > 0; off >>= 1) s += __shfl_xor(s, off, 32);
  if (lane == 0) x2[row] = s;
}

// ------- fused distance GEMM (f32 WMMA) + per-row top-34 selection -------
__global__ __launch_bounds__(128) void dao_main(const float* __restrict__ X,
                                                const float* __restrict__ x2,
                                                float* __restrict__ aiK,
                                                float* __restrict__ dfK,
                                                float* __restrict__ frl,
                                                int*   __restrict__ idxq) {
  __shared__ float tile[kROWS][kCT + 4];
  __shared__ float sld[kROWS][kNSEG][kLIST];
  __shared__ int   sli[kROWS][kNSEG][kLIST];

  const int tid  = threadIdx.x;
  const int wave = tid >> 5;            // == column segment 0..3
  const int lane = tid & 31;
  const int m    = lane & 15;
  const int half = lane >> 4;
  const int rowbase = blockIdx.x * kROWS;

  // init selection lists: one (row, seg) slot per thread
  {
    int row = tid & 31, seg = tid >> 5;
    for (int t = 0; t < kLIST; ++t) {
      sld[row][seg][t] = __FLT_MAX__;
      sli[row][seg][t] = 0x7fffffff;
    }
  }

  // per-lane row norms for both accumulator halves (loop invariant)
  float xr0[8], xr1[8];
#pragma unroll
  for (int j = 0; j < 8; ++j) {
    xr0[j] = x2[rowbase + j + 8 * half];
    xr1[j] = x2[rowbase + 16 + j + 8 * half];
  }
  const float* pa0 = X + (size_t)(rowbase + m) * kD + 2 * half;
  const float* pa1 = X + (size_t)(rowbase + 16 + m) * kD + 2 * half;

  __syncthreads();

  for (int colbase = 0; colbase < kN; colbase += kCT) {
    const int cb = colbase + 16 * wave;
    const float* pb = X + (size_t)(cb + m) * kD + 2 * half;
    const float xc = x2[cb + m];
    if (colbase + kCT < kN)
      __builtin_prefetch(X + (size_t)(colbase + kCT + 16 * wave + m) * kD, 0, 1);

    v8f c0 = {}; v8f c1 = {};
#pragma unroll 8
    for (int k0 = 0; k0 < kD; k0 += 4) {
      v2f b  = *(const v2f*)(pb  + k0);
      v2f a0 = *(const v2f*)(pa0 + k0);
      v2f a1 = *(const v2f*)(pa1 + k0);
      c0 = __builtin_amdgcn_wmma_f32_16x16x4_f32(false, a0, false, b,
                                                 (short)0, c0, false, false);
      c1 = __builtin_amdgcn_wmma_f32_16x16x4_f32(false, a1, false, b,
                                                 (short)0, c1, false, false);
    }

    // dot products -> clamped sqrt distances -> LDS tile
#pragma unroll
    for (int j = 0; j < 8; ++j) {
      float d2 = xr0[j] + xc - 2.f * c0[j];
      d2 = fmaxf(d2, 0.f);
      tile[j + 8 * half][16 * wave + m] = d2 > 0.f ? sqrtf(d2) : 0.f;
      float e2 = xr1[j] + xc - 2.f * c1[j];
      e2 = fmaxf(e2, 0.f);
      tile[16 + j + 8 * half][16 * wave + m] = e2 > 0.f ? sqrtf(e2) : 0.f;
    }
    __syncthreads();

    // incremental top-34: slot (row = lane, seg = wave) scans 16 columns
    {
      const int row = lane, seg = wave;
      float lmax = sld[row][seg][kLIST - 1];
      for (int cc = 0; cc < 16; ++cc) {
        float dv = tile[row][seg * 16 + cc];
        if (dv < lmax) {
          int gc = colbase + seg * 16 + cc;
          int k = kLIST - 1;
          while (k > 0 && sld[row][seg][k - 1] > dv) {
            sld[row][seg][k] = sld[row][seg][k - 1];
            sli[row][seg][k] = sli[row][seg][k - 1];
            --k;
          }
          sld[row][seg][k] = dv;
          sli[row][seg][k] = gc;
          lmax = sld[row][seg][kLIST - 1];
        }
      }
    }
    __syncthreads();
  }

  // merge the 4 sorted segment lists per row; one lane per row
  if (tid < kROWS) {
    const int row = tid;
    const int i = rowbase + row;
    int p[kNSEG] = {0, 0, 0, 0};
    float sv[kLIST]; int sc[kLIST];
    for (int t = 0; t < kLIST; ++t) {
      float best = __FLT_MAX__; int bc = 0x7fffffff; int bg = 0;
#pragma unroll
      for (int g = 0; g < kNSEG; ++g) {
        if (p[g] < kLIST) {
          float dv = sld[row][g][p[g]];
          int   cv = sli[row][g][p[g]];
          if (dv < best || (dv == best && cv < bc)) { best = dv; bc = cv; bg = g; }
        }
      }
      sv[t] = best; sc[t] = bc; p[bg]++;
    }
    // sv[0] == 0 (self). a_full[i,K] = sv[K]; a[i,K] = sv[K+1].
    const float sK = sv[kK];
    const float aK = sv[kK + 1];
    float sum = logf(kEPS);                      // log(0/sK + eps)
    for (int j = 1; j < kK; ++j) sum += logf(sv[j] / sK + kEPS);
    frl[i] = -(float)kK / sum;                   // full_rank_lids
    dfK[i] = sK;
    aiK[i] = aK;
    // reference quirk: removed-diagonal coordinates used as row indices
    for (int j = 0; j < kK; ++j) {
      int c = sc[j + 1];
      idxq[i * kK + j] = c - (c > i ? 1 : 0);
    }
  }
}

// ---------------- final gather + score ----------------
__global__ __launch_bounds__(256) void dao_scores(const float* __restrict__ aiK,
                                                  const float* __restrict__ dfK,
                                                  const float* __restrict__ frl,
                                                  const int*   __restrict__ idxq,
                                                  float* __restrict__ out) {
  int i = blockIdx.x * blockDim.x + threadIdx.x;
  if (i >= kN) return;
  const float ai = aiK[i];
  float s = 0.f;
  for (int j = 0; j < kK; ++j) {
    int q = idxq[i * kK + j];
    s += frl[q] * logf(ai / dfK[q]);
  }
  float v = s * (1.0f / kK);
  if (__builtin_isnan(v)) v = 1000.f;
  else if (__builtin_isinf(v)) v = v > 0.f ? 1000.f : 0.f;
  out[i] = v;
}

extern "C" void kernel_launch(void* const* d_in, const int* in_sizes, int n_in,
                              void* d_out, int out_size, void* d_ws, size_t ws_size,
                              hipStream_t stream) {
  const float* X = (const float*)d_in[0];
  float* x2  = (float*)d_ws;
  float* aiK = x2 + kN;
  float* dfK = aiK + kN;
  float* frl = dfK + kN;
  int*   idxq = (int*)(frl + kN);

  dao_norms<<<kN / 8, 256, 0, stream>>>(X, x2);
  dao_main<<<kN / kROWS, 128, 0, stream>>>(X, x2, aiK, dfK, frl, idxq);
  dao_scores<<<(kN + 255) / 256, 256, 0, stream>>>(aiK, dfK, frl, idxq,
                                                   (float*)d_out);
}